// RigidGroupProjector_43860206026933
// MI455X (gfx1250) — compile-verified
//
#include <hip/hip_runtime.h>

typedef float v2f __attribute__((ext_vector_type(2)));
typedef float v8f __attribute__((ext_vector_type(8)));

#define B_SZ  128
#define N_SZ  65536
#define NG_SZ 4096

// Jacobi rotation on symmetric Smat (full 3x3 storage), accumulating V <- V*J.
#define JROT(P, Q)                                                              \
  do {                                                                          \
    float apq = Smat[P][Q];                                                     \
    if (__builtin_fabsf(apq) > 1e-30f) {                                        \
      float tau = (Smat[Q][Q] - Smat[P][P]) / (2.0f * apq);                     \
      float tt  = __builtin_copysignf(1.0f, tau) /                              \
                  (__builtin_fabsf(tau) + __builtin_sqrtf(1.0f + tau * tau));   \
      float cc  = 1.0f / __builtin_sqrtf(1.0f + tt * tt);                       \
      float ss  = tt * cc;                                                      \
      _Pragma("unroll") for (int k = 0; k < 3; ++k) {                           \
        float skp = Smat[k][P], skq = Smat[k][Q];                               \
        Smat[k][P] = cc * skp - ss * skq;                                       \
        Smat[k][Q] = ss * skp + cc * skq;                                       \
      }                                                                         \
      _Pragma("unroll") for (int k = 0; k < 3; ++k) {                           \
        float spk = Smat[P][k], sqk = Smat[Q][k];                               \
        Smat[P][k] = cc * spk - ss * sqk;                                       \
        Smat[Q][k] = ss * spk + cc * sqk;                                       \
      }                                                                         \
      _Pragma("unroll") for (int k = 0; k < 3; ++k) {                           \
        float vkp = V[k][P], vkq = V[k][Q];                                     \
        V[k][P] = cc * vkp - ss * vkq;                                          \
        V[k][Q] = ss * vkp + cc * vkq;                                          \
      }                                                                         \
    }                                                                           \
  } while (0)

__global__ __launch_bounds__(256) void rigid_group_projector_kernel(
    const float* __restrict__ x,        // (B, N, 3)
    const float* __restrict__ tpl,      // (N, 3)
    const long long* __restrict__ grp,  // (NG, 16) int64
    float* __restrict__ out)            // (B, N, 3)
{
  // Per-wave R|t exchange buffer: 32 lanes x (9 R + 3 t), stride 13 (odd -> no bank conflicts)
  __shared__ float ldsRT[8][32 * 13];

  const int tid       = threadIdx.x;
  const int waveInBlk = tid >> 5;
  const int lane      = tid & 31;
  const int idx       = lane & 15;  // group-in-tile (and point/column id for WMMA)
  const int hi        = lane >> 4;  // batch-in-tile (and K-half / D-row-half for WMMA)

  const int waveId = blockIdx.x * 8 + waveInBlk;  // 16384 waves total
  const int gTile  = waveId & 255;                // 256 tiles of 16 groups
  const int bTile  = waveId >> 8;                 // 64 tiles of 2 batches
  const int g0     = gTile * 16;
  const int b0     = bTile * 2;

  float* const myLds = &ldsRT[waveInBlk][0];

  // ---- This lane's (batch, group) problem ----
  const int b = b0 + hi;
  const int g = g0 + idx;

  long long gi[16];
  {
    const long long* gp = grp + (long long)g * 16;
#pragma unroll
    for (int n = 0; n < 16; ++n) gi[n] = gp[n];
  }

  // ---- Single pass: sums and raw outer-product accumulation ----
  float Ss[3] = {0.f, 0.f, 0.f};
  float St[3] = {0.f, 0.f, 0.f};
  float Sst[3][3] = {};
  const size_t xb = (size_t)b * N_SZ;
#pragma unroll
  for (int n = 0; n < 16; ++n) {
    const float* sp = tpl + gi[n] * 3;
    const float* tp = x + (xb + (size_t)gi[n]) * 3;
    float s0 = sp[0], s1 = sp[1], s2 = sp[2];
    float t0 = tp[0], t1 = tp[1], t2 = tp[2];
    Ss[0] += s0; Ss[1] += s1; Ss[2] += s2;
    St[0] += t0; St[1] += t1; St[2] += t2;
    Sst[0][0] += s0 * t0; Sst[0][1] += s0 * t1; Sst[0][2] += s0 * t2;
    Sst[1][0] += s1 * t0; Sst[1][1] += s1 * t1; Sst[1][2] += s1 * t2;
    Sst[2][0] += s2 * t0; Sst[2][1] += s2 * t1; Sst[2][2] += s2 * t2;
  }
  const float inv16 = 1.0f / 16.0f;
  float ms[3] = {Ss[0] * inv16, Ss[1] * inv16, Ss[2] * inv16};
  float mt[3] = {St[0] * inv16, St[1] * inv16, St[2] * inv16};

  // H[i][j] = sum_n src_c[n][i] * tgt_c[n][j]
  float H[3][3];
#pragma unroll
  for (int i = 0; i < 3; ++i)
#pragma unroll
    for (int j = 0; j < 3; ++j) H[i][j] = Sst[i][j] - 16.0f * ms[i] * mt[j];

  // ---- S = H^T H, Jacobi eigendecomposition: S = V diag(lam) V^T ----
  float Smat[3][3];
#pragma unroll
  for (int i = 0; i < 3; ++i)
#pragma unroll
    for (int j = 0; j < 3; ++j) {
      float acc = 0.f;
#pragma unroll
      for (int k = 0; k < 3; ++k) acc += H[k][i] * H[k][j];
      Smat[i][j] = acc;
    }
  float V[3][3] = {{1.f, 0.f, 0.f}, {0.f, 1.f, 0.f}, {0.f, 0.f, 1.f}};
#pragma unroll
  for (int sweep = 0; sweep < 6; ++sweep) {
    JROT(0, 1);
    JROT(0, 2);
    JROT(1, 2);
  }

  float sg0 = __builtin_sqrtf(__builtin_fmaxf(Smat[0][0], 0.f));
  float sg1 = __builtin_sqrtf(__builtin_fmaxf(Smat[1][1], 0.f));
  float sg2 = __builtin_sqrtf(__builtin_fmaxf(Smat[2][2], 0.f));
  int mi = 0;
  float mv = sg0;
  if (sg1 < mv) { mv = sg1; mi = 1; }
  if (sg2 < mv) { mv = sg2; mi = 2; }

  float detH = H[0][0] * (H[1][1] * H[2][2] - H[1][2] * H[2][1]) -
               H[0][1] * (H[1][0] * H[2][2] - H[1][2] * H[2][0]) +
               H[0][2] * (H[1][0] * H[2][1] - H[1][1] * H[2][0]);
  float sgn = (detH < 0.f) ? -1.f : 1.f;

  const float tiny = 1e-20f;
  float d0 = ((mi == 0) ? sgn : 1.f) / __builtin_fmaxf(sg0, tiny);
  float d1 = ((mi == 1) ? sgn : 1.f) / __builtin_fmaxf(sg1, tiny);
  float d2 = ((mi == 2) ? sgn : 1.f) / __builtin_fmaxf(sg2, tiny);

  // M = V diag(d) V^T ; R = M H^T  (== V_fixed U^T of the reference)
  float M[3][3];
#pragma unroll
  for (int i = 0; i < 3; ++i)
#pragma unroll
    for (int j = 0; j < 3; ++j)
      M[i][j] = d0 * V[i][0] * V[j][0] + d1 * V[i][1] * V[j][1] + d2 * V[i][2] * V[j][2];

  float Rm[3][3];
#pragma unroll
  for (int i = 0; i < 3; ++i)
#pragma unroll
    for (int j = 0; j < 3; ++j) {
      float acc = 0.f;
#pragma unroll
      for (int k = 0; k < 3; ++k) acc += M[i][k] * H[j][k];  // M * H^T
      Rm[i][j] = acc;
    }
  float tv[3];
#pragma unroll
  for (int i = 0; i < 3; ++i)
    tv[i] = mt[i] - (Rm[i][0] * ms[0] + Rm[i][1] * ms[1] + Rm[i][2] * ms[2]);

  // ---- publish R|t to wave-local LDS ----
  {
    float* row = myLds + lane * 13;
#pragma unroll
    for (int i = 0; i < 3; ++i)
#pragma unroll
      for (int j = 0; j < 3; ++j) row[i * 3 + j] = Rm[i][j];
    row[9] = tv[0]; row[10] = tv[1]; row[11] = tv[2];
  }
  __syncthreads();

  // ---- WMMA projection: one v_wmma_f32_16x16x4_f32 per (group, batch-pair) ----
  // A(16x4): rows 0..2 = [R|t] of batch b0, rows 8..10 = [R|t] of batch b0+1.
  //   Lane layout (ISA): lane = (hi,idx); a.x = A[idx][2*hi], a.y = A[idx][2*hi+1].
  // B(4x16): col n = [sx, sy, sz, 1] of template point n of the group.
  //   Lane layout: b.x = B[2*hi][idx], b.y = B[2*hi+1][idx].
  // D(16x16): component v at lane (hi,idx) = D[v + 8*hi][idx]
  //   -> components 0..2 = (x,y,z) of point idx for batch b0+hi.
  //
  // Branchless fragment construction: clamped row index + validity-mask multiply
  // for A; one uniform b96 template load + selects for B. No EXEC churn.
  const int   pSel   = (idx >> 3) & 1;           // which batch's R|t this lane carries
  const int   mRow   = idx & 7;                  // A-row within that batch's 3 rows
  const int   mC     = (mRow < 3) ? mRow : 2;    // clamped (keeps LDS reads in-row)
  const float aValid = (mRow < 3) ? 1.0f : 0.0f; // rows 3..7 contribute nothing we read
  const int   offx   = hi ? (mC * 3 + 2) : (mC * 3);      // K=2*hi   column
  const int   offy   = hi ? (9 + mC)     : (mC * 3 + 1);  // K=2*hi+1 column
  const int   ldsRow0 = 16 * pSel;               // lane row base for this lane's source

#pragma unroll
  for (int q = 0; q < 16; ++q) {
    const float* rt = myLds + (ldsRow0 + q) * 13;
    v2f a;
    a.x = rt[offx] * aValid;
    a.y = rt[offy] * aValid;

    const long long pi = grp[(long long)(g0 + q) * 16 + idx];
    const float* sp = tpl + pi * 3;
    float s0 = sp[0], s1 = sp[1], s2 = sp[2];  // one b96 load, both halves
    v2f bm;
    bm.x = hi ? s2 : s0;
    bm.y = hi ? 1.0f : s1;

    v8f c = {0.f, 0.f, 0.f, 0.f, 0.f, 0.f, 0.f, 0.f};
    v8f d = __builtin_amdgcn_wmma_f32_16x16x4_f32(
        /*neg_a=*/false, a, /*neg_b=*/false, bm,
        /*c_mod=*/(short)0, c, /*reuse_a=*/false, /*reuse_b=*/false);

    float* o = out + ((size_t)(b0 + hi) * N_SZ + (size_t)pi) * 3;
    o[0] = d[0];
    o[1] = d[1];
    o[2] = d[2];
  }
}

extern "C" void kernel_launch(void* const* d_in, const int* in_sizes, int n_in,
                              void* d_out, int out_size, void* d_ws, size_t ws_size,
                              hipStream_t stream) {
  const float* x        = (const float*)d_in[0];      // (128, 65536, 3) f32
  const float* tpl      = (const float*)d_in[1];      // (65536, 3) f32
  const long long* grpp = (const long long*)d_in[2];  // (4096, 16) i64
  float* out            = (float*)d_out;              // (128, 65536, 3) f32

  // 16384 waves = 64 batch-tiles(2) x 256 group-tiles(16); 8 waves/block
  dim3 grid(2048), block(256);
  hipLaunchKernelGGL(rigid_group_projector_kernel, grid, block, 0, stream,
                     x, tpl, grpp, out);
}